// LightGCN_multi_43198781063356
// MI455X (gfx1250) — compile-verified
//
#include <hip/hip_runtime.h>
#include <hip/hip_bf16.h>
#include <math.h>

// ---------------------------------------------------------------------------
// LightGCN multi-batch forward for MI455X (gfx1250, wave32).
// Scatter/gather segment-sums use hardware GLOBAL_ATOMIC_ADD_F32 (no-return).
// Dense projection uses V_WMMA_F32_16X16X4_F32 (full f32 precision).
// ---------------------------------------------------------------------------

#define NCELL 50000
#define NGENE 2000
#define DFEAT 64
#define NEDGE 1000000
#define NADJ  800000
#define NPRED 200000

typedef __attribute__((ext_vector_type(2))) float v2f;
typedef __attribute__((ext_vector_type(8))) float v8f;

__device__ __forceinline__ void atomAddF(float* p, float v) {
    // Relaxed agent-scope fp32 atomic add; discarded return -> non-returning
    // GLOBAL_ATOMIC_ADD_F32 (tracked on STOREcnt), no CAS loop.
    __hip_atomic_fetch_add(p, v, __ATOMIC_RELAXED, __HIP_MEMORY_SCOPE_AGENT);
}

// ---- degree^{-1/2} --------------------------------------------------------
__global__ void k_deg_count(const int* __restrict__ idx, float* __restrict__ deg, int n) {
    int t = blockIdx.x * blockDim.x + threadIdx.x;
    if (t < n) atomAddF(&deg[idx[t]], 1.0f);
}

__global__ void k_invsqrt(float* __restrict__ d, int n) {
    int t = blockIdx.x * blockDim.x + threadIdx.x;
    if (t < n) {
        float x = d[t];
        d[t] = (x > 0.0f) ? rsqrtf(x) : 0.0f;
    }
}

// ---- elementwise helpers --------------------------------------------------
__global__ void k_scale_copy(const float* __restrict__ in, float* __restrict__ out,
                             float a, int n) {
    int t = blockIdx.x * blockDim.x + threadIdx.x;
    if (t < n) out[t] = a * in[t];
}

// nf = nf*selfscale; hid += w*nf
__global__ void k_commit(float* __restrict__ nf, float* __restrict__ hid,
                         float selfscale, float w, int n) {
    int t = blockIdx.x * blockDim.x + threadIdx.x;
    if (t < n) {
        float x = nf[t] * selfscale;
        nf[t] = x;
        hid[t] += w * x;
    }
}

// ---- LightGraphConv scatter: out[dst] += feat[src]*cj[src]*ci[dst] -------
// One thread per (edge, 4-float chunk): float4 (B128) gather, 4 f32 atomics.
__global__ void k_lgc_scatter(const float* __restrict__ feat, const int* __restrict__ sidx,
                              const int* __restrict__ didx, const float* __restrict__ cj,
                              const float* __restrict__ ci, float* __restrict__ out, int E) {
    int t = blockIdx.x * blockDim.x + threadIdx.x;
    int e = t >> 4;
    if (e >= E) return;
    int d = (t & 15) << 2;
    int s = sidx[e], q = didx[e];
    float sc = cj[s] * ci[q];
    float4 v = *(const float4*)(feat + (size_t)s * DFEAT + d);
    float* o = out + (size_t)q * DFEAT + d;
    atomAddF(o + 0, sc * v.x);
    atomAddF(o + 1, sc * v.y);
    atomAddF(o + 2, sc * v.z);
    atomAddF(o + 3, sc * v.w);
}

// ---- sparse.mm scatter: out[row] += val*feat[col] -------------------------
__global__ void k_spmm_scatter(const float* __restrict__ feat, const int* __restrict__ row,
                               const int* __restrict__ col, const float* __restrict__ val,
                               float* __restrict__ out, int E) {
    int t = blockIdx.x * blockDim.x + threadIdx.x;
    int e = t >> 4;
    if (e >= E) return;
    int d = (t & 15) << 2;
    float sc = val[e];
    float4 v = *(const float4*)(feat + (size_t)col[e] * DFEAT + d);
    float* o = out + (size_t)row[e] * DFEAT + d;
    atomAddF(o + 0, sc * v.x);
    atomAddF(o + 1, sc * v.y);
    atomAddF(o + 2, sc * v.z);
    atomAddF(o + 3, sc * v.w);
}

// ---- dense projection via V_WMMA_F32_16X16X4_F32 --------------------------
// h[m][n] = sum_k z[m][k] * emb_w[n][k] + emb_b[n] + emb_w[n][64+batch]
// One wave (32 threads) per 16x16 output tile; 16 K-steps of 4.
// A(16x4 f32): lane = M (0..15 twice); lanes 0-15 hold K={0,1}, lanes 16-31 K={2,3}.
// B(4x16 f32): lane = N; same K-half split. D: VGPR r -> row r (+8 for hi lanes).
__global__ void k_gemm_bias(const float* __restrict__ z, const float* __restrict__ embw,
                            const float* __restrict__ embb, float* __restrict__ h,
                            int batch) {
    int lane = threadIdx.x;
    int half = lane >> 4;      // 0: low K pair, 1: high K pair
    int lm   = lane & 15;      // M index for A, N index for B/D
    int m0 = blockIdx.x * 16;
    int n0 = blockIdx.y * 16;
    int n  = n0 + lm;
    float bias = embb[n] + embw[(size_t)n * 66 + 64 + batch];
#if defined(__gfx1250__) && __has_builtin(__builtin_amdgcn_wmma_f32_16x16x4_f32)
    const float* za = z + (size_t)(m0 + lm) * DFEAT;     // A row m = lm
    const float* wb = embw + (size_t)n * 66;             // B col n = lm (row-major W)
    v8f acc = {};
#pragma unroll
    for (int kb = 0; kb < DFEAT; kb += 4) {
        v2f a, b;
        a.x = za[kb + 2 * half + 0];
        a.y = za[kb + 2 * half + 1];
        b.x = wb[kb + 2 * half + 0];
        b.y = wb[kb + 2 * half + 1];
        acc = __builtin_amdgcn_wmma_f32_16x16x4_f32(
            /*neg_a=*/false, a, /*neg_b=*/false, b,
            /*c_mod=*/(short)0, acc, /*reuse_a=*/false, /*reuse_b=*/false);
    }
#pragma unroll
    for (int r = 0; r < 8; ++r)
        h[(size_t)(m0 + half * 8 + r) * DFEAT + n] = acc[r] + bias;
#else
    // Scalar fallback (compiles everywhere; gfx1250 should take the WMMA path).
    const float* wr = embw + (size_t)n * 66;
    for (int r = 0; r < 8; ++r) {
        const float* zr = z + (size_t)(m0 + half * 8 + r) * DFEAT;
        float s = 0.0f;
        for (int k = 0; k < DFEAT; ++k) s += zr[k] * wr[k];
        h[(size_t)(m0 + half * 8 + r) * DFEAT + n] = s + bias;
    }
#endif
}

// ---- BatchNorm (train stats) + ELU ----------------------------------------
__global__ void k_bn_partial(const float* __restrict__ h, float* __restrict__ stats, int M) {
    __shared__ float red[512];
    int d   = threadIdx.x & 63;
    int sub = threadIdx.x >> 6;   // 4 row-lanes per feature
    float s = 0.0f, q = 0.0f;
    for (int m = blockIdx.x * 4 + sub; m < M; m += gridDim.x * 4) {
        float x = h[(size_t)m * DFEAT + d];
        s += x;
        q += x * x;
    }
    red[threadIdx.x]       = s;
    red[256 + threadIdx.x] = q;
    __syncthreads();
    if (sub == 0) {
        float ss = red[d] + red[64 + d] + red[128 + d] + red[192 + d];
        float qq = red[256 + d] + red[320 + d] + red[384 + d] + red[448 + d];
        atomAddF(&stats[d], ss);
        atomAddF(&stats[64 + d], qq);
    }
}

__global__ void k_bn_finalize(float* __restrict__ stats, const float* __restrict__ gamma,
                              const float* __restrict__ beta, float invM) {
    int d = threadIdx.x;   // 64 threads
    float mu  = stats[d] * invM;
    float var = stats[64 + d] * invM - mu * mu;
    float sc  = rsqrtf(var + 1e-5f) * gamma[d];
    stats[128 + d] = sc;
    stats[192 + d] = beta[d] - mu * sc;
}

__global__ void k_bn_elu(float* __restrict__ h, const float* __restrict__ stats, int n) {
    int t = blockIdx.x * blockDim.x + threadIdx.x;
    if (t < n) {
        int d = t & 63;
        float x = h[t] * stats[128 + d] + stats[192 + d];
        h[t] = (x > 0.0f) ? x : (expf(x) - 1.0f);
    }
}

// ---- DotDecoder: sigmoid(h[u] . g[v]) -------------------------------------
__global__ void k_decode(const float* __restrict__ h, const float* __restrict__ g,
                         const int* __restrict__ u, const int* __restrict__ v,
                         float* __restrict__ out, int n) {
    int t = blockIdx.x * blockDim.x + threadIdx.x;
    if (t >= n) return;
    const float4* hr = (const float4*)(h + (size_t)u[t] * DFEAT);
    const float4* gr = (const float4*)(g + (size_t)v[t] * DFEAT);
    float s = 0.0f;
#pragma unroll
    for (int k = 0; k < 16; ++k) {
        float4 a = hr[k], b = gr[k];
        s += a.x * b.x + a.y * b.y + a.z * b.z + a.w * b.w;
    }
    out[t] = 1.0f / (1.0f + expf(-s));
}

// ---------------------------------------------------------------------------
extern "C" void kernel_launch(void* const* d_in, const int* in_sizes, int n_in,
                              void* d_out, int out_size, void* d_ws, size_t ws_size,
                              hipStream_t stream) {
    (void)in_sizes; (void)n_in; (void)out_size; (void)ws_size;

    const float* gene_feature    = (const float*)d_in[0];
    const float* cell_feature[2] = {(const float*)d_in[1], (const float*)d_in[2]};
    const float* emb_w           = (const float*)d_in[3];
    const float* emb_b           = (const float*)d_in[4];
    const float* bn_gamma        = (const float*)d_in[5];
    const float* bn_beta         = (const float*)d_in[6];
    const float* adj_val[2]      = {(const float*)d_in[7],  (const float*)d_in[8]};
    const int*   eg_src[2]       = {(const int*)d_in[9],    (const int*)d_in[17]};
    const int*   eg_dst[2]       = {(const int*)d_in[10],   (const int*)d_in[18]};
    const int*   adj_row[2]      = {(const int*)d_in[11],   (const int*)d_in[19]};
    const int*   adj_col[2]      = {(const int*)d_in[12],   (const int*)d_in[20]};
    const int*   pos_u[2]        = {(const int*)d_in[13],   (const int*)d_in[21]};
    const int*   pos_v[2]        = {(const int*)d_in[14],   (const int*)d_in[22]};
    const int*   neg_u[2]        = {(const int*)d_in[15],   (const int*)d_in[23]};
    const int*   neg_v[2]        = {(const int*)d_in[16],   (const int*)d_in[24]};
    float* out = (float*)d_out;

    // ---- workspace layout (floats) ----
    float* ws  = (float*)d_ws;
    size_t off = 0;
    auto alloc = [&](size_t n) { float* p = ws + off; off += n; return p; };
    float* c_cell[2] = {alloc(50048), alloc(50048)};
    float* c_gene[2] = {alloc(2048),  alloc(2048)};
    const size_t CF = (size_t)NCELL * DFEAT;     // 3.2M floats
    const size_t GF = (size_t)NGENE * DFEAT;     // 128K floats
    float* bufA[2] = {alloc(CF), alloc(CF)};     // layer-0 new cell feats / z / h
    float* bufB[2] = {alloc(CF), alloc(CF)};     // layer-1 new cell feats / z2
    float* uh[2]   = {alloc(CF), alloc(CF)};     // u_hid accumulators
    float* gA      = alloc(GF);                  // layer-0 new gene feats
    float* gB      = alloc(GF);                  // layer-1 new gene feats
    float* gh      = alloc(GF);                  // i_hid accumulator
    float* stats   = alloc(256);                 // [sum|sumsq|scale|shift] x 64

    const float W13 = 1.0f / 3.0f;
    const int T = 256;
    const int gEdge  = (NEDGE + T - 1) / T;            // 1 thread / edge
    const int gScat  = (NEDGE * 16 + T - 1) / T;       // 16 threads / edge
    const int gAdj   = (NADJ * 16 + T - 1) / T;
    const int gCellF = ((int)CF + T - 1) / T;
    const int gGeneF = ((int)GF + T - 1) / T;
    const int gPred  = (NPRED + T - 1) / T;

    // ---- degree^{-1/2} per node ----
    for (int i = 0; i < 2; ++i) {
        hipMemsetAsync(c_cell[i], 0, (size_t)NCELL * 4, stream);
        hipMemsetAsync(c_gene[i], 0, (size_t)NGENE * 4, stream);
        k_deg_count<<<gEdge, T, 0, stream>>>(eg_src[i], c_cell[i], NEDGE);
        k_deg_count<<<gEdge, T, 0, stream>>>(eg_dst[i], c_gene[i], NEDGE);
        k_invsqrt<<<(NCELL + T - 1) / T, T, 0, stream>>>(c_cell[i], NCELL);
        k_invsqrt<<<(NGENE + T - 1) / T, T, 0, stream>>>(c_gene[i], NGENE);
    }

    // ---- init hidden accumulators: hid = W0 * feats ----
    k_scale_copy<<<gCellF, T, 0, stream>>>(cell_feature[0], uh[0], W13, (int)CF);
    k_scale_copy<<<gCellF, T, 0, stream>>>(cell_feature[1], uh[1], W13, (int)CF);
    k_scale_copy<<<gGeneF, T, 0, stream>>>(gene_feature, gh, W13, (int)GF);

    // ---- 2 LightGCN layers ----
    const float* ufcur[2] = {cell_feature[0], cell_feature[1]};
    const float* gfcur    = gene_feature;
    float* cellPing[2][2] = {{bufA[0], bufA[1]}, {bufB[0], bufB[1]}};
    float* genePing[2]    = {gA, gB};
    for (int l = 0; l < 2; ++l) {
        float* ufn[2] = {cellPing[l][0], cellPing[l][1]};
        float* gfn    = genePing[l];
        hipMemsetAsync(gfn, 0, GF * 4, stream);
        hipMemsetAsync(ufn[0], 0, CF * 4, stream);
        hipMemsetAsync(ufn[1], 0, CF * 4, stream);
        // gene update over both relations (ci*cj folded into the message)
        for (int i = 0; i < 2; ++i)
            k_lgc_scatter<<<gScat, T, 0, stream>>>(ufcur[i], eg_src[i], eg_dst[i],
                                                   c_cell[i], c_gene[i], gfn, NEDGE);
        // per-batch cell update via reverse edges
        for (int i = 0; i < 2; ++i)
            k_lgc_scatter<<<gScat, T, 0, stream>>>(gfcur, eg_dst[i], eg_src[i],
                                                   c_gene[i], c_cell[i], ufn[i], NEDGE);
        // commit: gene mean over 2 relations (x0.5); accumulate hiddens
        k_commit<<<gGeneF, T, 0, stream>>>(gfn, gh, 0.5f, W13, (int)GF);
        k_commit<<<gCellF, T, 0, stream>>>(ufn[0], uh[0], 1.0f, W13, (int)CF);
        k_commit<<<gCellF, T, 0, stream>>>(ufn[1], uh[1], 1.0f, W13, (int)CF);
        ufcur[0] = ufn[0]; ufcur[1] = ufn[1]; gfcur = gfn;
    }

    // ---- per-batch: spmm x2, WMMA projection, BN(train)+ELU ----
    float* hcell[2];
    for (int i = 0; i < 2; ++i) {
        float* z1 = bufA[i];   // dead after layers -> spmm #1 output, later h
        float* z2 = bufB[i];   // last ufeats, dead -> spmm #2 output
        hipMemsetAsync(z1, 0, CF * 4, stream);
        k_spmm_scatter<<<gAdj, T, 0, stream>>>(uh[i], adj_row[i], adj_col[i],
                                               adj_val[i], z1, NADJ);
        hipMemsetAsync(z2, 0, CF * 4, stream);
        k_spmm_scatter<<<gAdj, T, 0, stream>>>(z1, adj_row[i], adj_col[i],
                                               adj_val[i], z2, NADJ);
        float* h = z1;  // overwrite z1 with projection result
        k_gemm_bias<<<dim3(NCELL / 16, DFEAT / 16), 32, 0, stream>>>(z2, emb_w, emb_b, h, i);
        hipMemsetAsync(stats, 0, 128 * 4, stream);
        k_bn_partial<<<256, 256, 0, stream>>>(h, stats, NCELL);
        k_bn_finalize<<<1, 64, 0, stream>>>(stats, bn_gamma, bn_beta, 1.0f / NCELL);
        k_bn_elu<<<gCellF, T, 0, stream>>>(h, stats, (int)CF);
        hcell[i] = h;
    }

    // ---- decoder: sigmoid(h[u] . i_hid[v]); concat pos1,pos2,neg1,neg2 ----
    k_decode<<<gPred, T, 0, stream>>>(hcell[0], gh, pos_u[0], pos_v[0], out + 0 * NPRED, NPRED);
    k_decode<<<gPred, T, 0, stream>>>(hcell[1], gh, pos_u[1], pos_v[1], out + 1 * NPRED, NPRED);
    k_decode<<<gPred, T, 0, stream>>>(hcell[0], gh, neg_u[0], neg_v[0], out + 2 * NPRED, NPRED);
    k_decode<<<gPred, T, 0, stream>>>(hcell[1], gh, neg_u[1], neg_v[1], out + 3 * NPRED, NPRED);
}